// MeanHinAggregator_50294067036664
// MI455X (gfx1250) — compile-verified
//
#include <hip/hip_runtime.h>
#include <hip/hip_bf16.h>

typedef __attribute__((ext_vector_type(2))) float v2f;
typedef __attribute__((ext_vector_type(4))) float v4f;
typedef __attribute__((ext_vector_type(8))) float v8f;

#define B_DIM 1024
#define H_DIM 10
#define N_DIM 32
#define F_DIM 128
#define HALFD 128
#define M_TOT (B_DIM * H_DIM)   // 10240 rows of (b,h)
#define PAD   132               // 128 + 4 : bank-conflict-free stride

// One K=128 GEMM pass: acc[4] (+)= A(16x128 from LDS) * W[:, nbase:nbase+64]
// A-fragment (ISA 7.12.2, 32-bit A 16x4): lanes 0-15 = M, VGPR0/1 = K0/K1;
// lanes 16-31 = K2/K3.  B mirrored: lane%16 = N, lane/16 selects K pair.
__device__ __forceinline__ void gemm_pass(const float* __restrict__ sA,
                                          const float* __restrict__ wp,
                                          int am, int ak, int bn, int bk,
                                          int nbase, v8f acc[4])
{
#pragma unroll 4
    for (int kk = 0; kk < 32; ++kk) {
        const int kl = kk * 4;
        const v2f a = *reinterpret_cast<const v2f*>(&sA[am * PAD + kl + ak]);
#pragma unroll
        for (int t = 0; t < 4; ++t) {
            const int n0 = nbase + t * 16;
            v2f bf;
            bf.x = wp[(kl + bk)     * HALFD + n0 + bn];
            bf.y = wp[(kl + bk + 1) * HALFD + n0 + bn];
            // (neg_a, A, neg_b, B, c_mod, C, reuse_a, reuse_b)
            acc[t] = __builtin_amdgcn_wmma_f32_16x16x4_f32(
                false, a, false, bf, (short)0, acc[t], false, false);
        }
    }
}

__global__ __launch_bounds__(128) void hin_agg_kernel(
    const float* __restrict__ x_self,
    const float* __restrict__ xn0,
    const float* __restrict__ xn1,
    const float* __restrict__ w_self,
    const float* __restrict__ w0,
    const float* __restrict__ w1,
    const float* __restrict__ bias,
    float* __restrict__ out)
{
    __shared__ float sSelf[16 * PAD];
    __shared__ float sN0[16 * PAD];
    __shared__ float sN1[16 * PAD];

    const int tid  = threadIdx.x;
    const int wave = tid >> 5;
    const int lane = tid & 31;
    const int m0   = blockIdx.x * 16;   // first (b,h) row of this block

    // ---------------- Phase 1a: stage x_self tile (16 x 128) into LDS --------
    for (int i = tid; i < 16 * 32; i += 128) {
        const int row = i >> 5;
        const int cg  = i & 31;
        const v4f v = *reinterpret_cast<const v4f*>(
            x_self + (size_t)(m0 + row) * F_DIM + cg * 4);
        *reinterpret_cast<v4f*>(&sSelf[row * PAD + cg * 4]) = v;
    }

    // ---------------- Phase 1b: mean over the neighbour axis ------------------
    // Each wave owns 4 rows; each lane owns 4 contiguous columns (float4).
    // 335 MB streamed exactly once -> non-temporal loads so the shared
    // weights/bias stay resident in L2 for all 640 blocks.
    {
        const int cg = lane;            // column group: cols 4*cg .. 4*cg+3
        const float inv = 1.0f / (float)N_DIM;
        for (int rr = 0; rr < 4; ++rr) {
            const int row = wave * 4 + rr;
            const size_t base = (size_t)(m0 + row) * N_DIM * F_DIM + (size_t)cg * 4;
            v4f a0 = (v4f)(0.0f);
            v4f a1 = (v4f)(0.0f);
#pragma unroll 8
            for (int n = 0; n < N_DIM; ++n) {
                const v4f u = __builtin_nontemporal_load(
                    reinterpret_cast<const v4f*>(xn0 + base + (size_t)n * F_DIM));
                const v4f w = __builtin_nontemporal_load(
                    reinterpret_cast<const v4f*>(xn1 + base + (size_t)n * F_DIM));
                a0 += u;
                a1 += w;
            }
            a0 *= inv;
            a1 *= inv;
            *reinterpret_cast<v4f*>(&sN0[row * PAD + cg * 4]) = a0;
            *reinterpret_cast<v4f*>(&sN1[row * PAD + cg * 4]) = a1;
        }
    }

    __syncthreads();

    // ---------------- Phase 2: WMMA f32 16x16x4 GEMMs -------------------------
    // waves 0,1 : self half  (A = sSelf,      K = 128, B = w_self)
    // waves 2,3 : neigh half (A = [sN0|sN1],  K = 256, B = [w0;w1])
    const int half  = wave >> 1;         // 0 = self cols, 1 = neigh cols
    const int nbase = (wave & 1) * 64;   // column base inside the half

    const int am = lane & 15;
    const int ak = (lane >> 4) * 2;
    const int bn = lane & 15;
    const int bk = (lane >> 4) * 2;

    v8f acc[4];
#pragma unroll
    for (int t = 0; t < 4; ++t)
#pragma unroll
        for (int r = 0; r < 8; ++r) acc[t][r] = 0.0f;

    // Wave-uniform branch: each side is a straight-line loop with
    // loop-invariant base pointers (no per-iteration exec juggling).
    if (half == 0) {
        gemm_pass(sSelf, w_self, am, ak, bn, bk, nbase, acc);
    } else {
        gemm_pass(sN0, w0, am, ak, bn, bk, nbase, acc);
        gemm_pass(sN1, w1, am, ak, bn, bk, nbase, acc);
    }

    // ---------------- Epilogue: scale (1/NR for neigh), +bias, ReLU, store ----
    // C/D layout: VGPR r -> M = r + 8*(lane/16), lane%16 -> N
    const int rowhi = (lane >> 4) * 8;
    const float scale = half ? 0.5f : 1.0f;
#pragma unroll
    for (int t = 0; t < 4; ++t) {
        const int col = half * HALFD + nbase + t * 16 + (lane & 15);
        const float bv = bias[col];
#pragma unroll
        for (int r = 0; r < 8; ++r) {
            const int row = m0 + r + rowhi;
            const float v = acc[t][r] * scale + bv;
            out[(size_t)row * (2 * HALFD) + col] = fmaxf(v, 0.0f);
        }
    }
}

extern "C" void kernel_launch(void* const* d_in, const int* in_sizes, int n_in,
                              void* d_out, int out_size, void* d_ws, size_t ws_size,
                              hipStream_t stream) {
    const float* x_self = (const float*)d_in[0];
    const float* xn0    = (const float*)d_in[1];
    const float* xn1    = (const float*)d_in[2];
    const float* w_self = (const float*)d_in[3];
    const float* w0     = (const float*)d_in[4];
    const float* w1     = (const float*)d_in[5];
    const float* bias   = (const float*)d_in[6];
    float* out = (float*)d_out;

    const int blocks = M_TOT / 16;  // 640
    hin_agg_kernel<<<blocks, 128, 0, stream>>>(x_self, xn0, xn1,
                                               w_self, w0, w1, bias, out);
}